// UnbiasedVICRegLoss_31361851195491
// MI455X (gfx1250) — compile-verified
//
#include <hip/hip_runtime.h>
#include <math.h>

typedef float v2f __attribute__((ext_vector_type(2)));
typedef float v8f __attribute__((ext_vector_type(8)));

#define N_ROWS 32768
#define DIM 512
#define HALF 16384
#define TILE 64
#define KCHUNK 64
#define KSPLIT 4
#define KSEG (HALF / KSPLIT)          // 4096 rows per K-segment

// Workspace layout (float elements):
#define GPART_SZ 262144               // 512*512 per partial Gram
// G partials: [mat(2)][ksplit(4)] -> 8 * 262144
#define REPR_OFF 2097152              // 2048 block partials
#define REPR_BLOCKS 2048
#define COV_OFF 2099200               // 64 block partials
#define COV_BLOCKS 64
// total: 2099264 floats (~8.4 MB) of d_ws

// ---------------------------------------------------------------------------
// Kernel 1: repr_loss partial sums: sum (a-b)^2 per block (deterministic).
// ---------------------------------------------------------------------------
__global__ __launch_bounds__(256) void repr_partial_kernel(
    const float* __restrict__ a, const float* __restrict__ b,
    float* __restrict__ partials) {
  __shared__ float red[256];
  const int tid = threadIdx.x;
  const size_t idx = (size_t)blockIdx.x * 256 + tid;
  const size_t stride = (size_t)REPR_BLOCKS * 256;
  const size_t total4 = (size_t)N_ROWS * DIM / 4;
  const float4* a4 = (const float4*)a;
  const float4* b4 = (const float4*)b;
  float s = 0.f;
  for (size_t i = idx; i < total4; i += stride) {
    float4 va = a4[i], vb = b4[i];
    float dx = va.x - vb.x, dy = va.y - vb.y;
    float dz = va.z - vb.z, dw = va.w - vb.w;
    s += dx * dx + dy * dy + dz * dz + dw * dw;
  }
  red[tid] = s;
  __syncthreads();
  for (int off = 128; off > 0; off >>= 1) {
    if (tid < off) red[tid] += red[tid + off];
    __syncthreads();
  }
  if (tid == 0) partials[blockIdx.x] = red[0];
}

// ---------------------------------------------------------------------------
// Kernel 2: partial Gram matrices G = Z^T Z over one K-segment (f32 WMMA).
// grid: x = 64 tiles (8x8 of 64x64), y = matrix (0/1), z = K-segment (0..3).
// Each block writes its own partial buffer -> no atomics, deterministic.
// ---------------------------------------------------------------------------
__global__ __launch_bounds__(256) void gram_kernel(
    const float* __restrict__ z_a, const float* __restrict__ z_b,
    const int* __restrict__ perm, float* __restrict__ ws) {
  __shared__ __attribute__((aligned(16))) float ldsA[TILE * TILE];
  __shared__ __attribute__((aligned(16))) float ldsB[TILE * TILE];

  const int tileId = blockIdx.x;      // 0..63
  const int matId  = blockIdx.y;      // 0 or 1
  const int ks     = blockIdx.z;      // 0..KSPLIT-1
  const int tm = tileId >> 3;         // output row-tile
  const int tn = tileId & 7;          // output col-tile
  float* __restrict__ G = ws + (size_t)(matId * KSPLIT + ks) * GPART_SZ;

  const int tid  = threadIdx.x;
  const int lane = tid & 31;
  const int wave = tid >> 5;
  const int ml = lane & 15;           // M (or N) index within 16
  const int kb = (lane >> 4) << 1;    // K sub-offset: 0 or 2

  // two 16x16 subtiles per wave out of the 4x4 grid (shared A fragment)
  const int s0 = wave * 2, s1 = wave * 2 + 1;
  const int sm0 = s0 >> 2, sn0 = s0 & 3;
  const int sm1 = s1 >> 2, sn1 = s1 & 3;
  v8f acc0 = {}; v8f acc1 = {};

  const int r  = tid >> 2;            // gathered row within chunk: 0..63
  const int c4 = (tid & 3) * 16;      // 16-float column segment
  const bool diag = (tm == tn);

  const int kbase = ks * KSEG;
  for (int k0 = kbase; k0 < kbase + KSEG; k0 += KCHUNK) {
    // Gather one row of the permuted set; stage 64-col slices into LDS.
    const int gidx = perm[matId * HALF + k0 + r];
    const float* src = (gidx < N_ROWS) ? (z_a + (size_t)gidx * DIM)
                                       : (z_b + (size_t)(gidx - N_ROWS) * DIM);
    {
      const float4* pa = (const float4*)(src + tm * TILE + c4);
      float4* da = (float4*)(ldsA + r * TILE + c4);
      da[0] = pa[0]; da[1] = pa[1]; da[2] = pa[2]; da[3] = pa[3];
      if (!diag) {
        const float4* pb = (const float4*)(src + tn * TILE + c4);
        float4* db = (float4*)(ldsB + r * TILE + c4);
        db[0] = pb[0]; db[1] = pb[1]; db[2] = pb[2]; db[3] = pb[3];
      }
    }
    __syncthreads();
    const float* __restrict__ A = ldsA;
    const float* __restrict__ B = diag ? ldsA : ldsB;
    for (int kk = 0; kk < KCHUNK; kk += 4) {
      const int kr0 = (kk + kb) * TILE;
      const int kr1 = kr0 + TILE;
      v2f a0, b0, b1;
      a0.x = A[kr0 + sm0 * 16 + ml]; a0.y = A[kr1 + sm0 * 16 + ml];
      b0.x = B[kr0 + sn0 * 16 + ml]; b0.y = B[kr1 + sn0 * 16 + ml];
      acc0 = __builtin_amdgcn_wmma_f32_16x16x4_f32(false, a0, false, b0,
                                                   (short)0, acc0, false, false);
      b1.x = B[kr0 + sn1 * 16 + ml]; b1.y = B[kr1 + sn1 * 16 + ml];
      acc1 = __builtin_amdgcn_wmma_f32_16x16x4_f32(false, a0, false, b1,
                                                   (short)0, acc1, false, false);
    }
    __syncthreads();
  }

  // Store D tiles: lane holds row = 16*sm + (lane>>4)*8 + r, col = 16*sn + ml.
  const int rowBase = tm * TILE;
  const int colBase = tn * TILE;
  const int radd = (lane >> 4) * 8;
#pragma unroll
  for (int rr = 0; rr < 8; ++rr) {
    const int row0 = rowBase + sm0 * 16 + radd + rr;
    G[(size_t)row0 * DIM + colBase + sn0 * 16 + ml] = acc0[rr];
    const int row1 = rowBase + sm1 * 16 + radd + rr;
    G[(size_t)row1 * DIM + colBase + sn1 * 16 + ml] = acc1[rr];
  }
}

// ---------------------------------------------------------------------------
// Kernel 3: cov_diff = (G1/denom - I)(G2/denom - I); per-block sum of squares.
// Sums the KSPLIT partial Gram buffers on load; uses symmetry of G1 for the
// WMMA A-operand (row-major feed, no transpose staging).
// ---------------------------------------------------------------------------
__global__ __launch_bounds__(256) void covdiff_kernel(
    const float* __restrict__ ws_ro, float* __restrict__ covPartials) {
  __shared__ __attribute__((aligned(16))) float ldsA[TILE * TILE];
  __shared__ __attribute__((aligned(16))) float ldsB[TILE * TILE];
  __shared__ float red[256];
  const float inv_denom = 1.0f / 16383.0f;

  const int tileId = blockIdx.x;
  const int tm = tileId >> 3, tn = tileId & 7;
  const float* __restrict__ G1p = ws_ro;                         // 4 partials
  const float* __restrict__ G2p = ws_ro + (size_t)KSPLIT * GPART_SZ;

  const int tid  = threadIdx.x;
  const int lane = tid & 31;
  const int wave = tid >> 5;
  const int ml = lane & 15;
  const int kb = (lane >> 4) << 1;
  const int s0 = wave * 2, s1 = wave * 2 + 1;
  const int sm0 = s0 >> 2, sn0 = s0 & 3;
  const int sm1 = s1 >> 2, sn1 = s1 & 3;
  v8f acc0 = {}; v8f acc1 = {};

  const int r  = tid >> 2;
  const int c4 = (tid & 3) * 16;

  for (int k0 = 0; k0 < DIM; k0 += KCHUNK) {
    const int krow = k0 + r;
#pragma unroll 4
    for (int j = 0; j < 16; ++j) {
      const int ca = tm * TILE + c4 + j;   // A via symmetry: M1[krow][ca]
      const int cb = tn * TILE + c4 + j;
      float va = 0.f, vb = 0.f;
#pragma unroll
      for (int p = 0; p < KSPLIT; ++p) {
        va += G1p[(size_t)p * GPART_SZ + (size_t)krow * DIM + ca];
        vb += G2p[(size_t)p * GPART_SZ + (size_t)krow * DIM + cb];
      }
      ldsA[r * TILE + c4 + j] = va * inv_denom - (krow == ca ? 1.0f : 0.0f);
      ldsB[r * TILE + c4 + j] = vb * inv_denom - (krow == cb ? 1.0f : 0.0f);
    }
    __syncthreads();
    for (int kk = 0; kk < KCHUNK; kk += 4) {
      const int kr0 = (kk + kb) * TILE;
      const int kr1 = kr0 + TILE;
      v2f a0, b0, a1, b1;
      a0.x = ldsA[kr0 + sm0 * 16 + ml]; a0.y = ldsA[kr1 + sm0 * 16 + ml];
      b0.x = ldsB[kr0 + sn0 * 16 + ml]; b0.y = ldsB[kr1 + sn0 * 16 + ml];
      acc0 = __builtin_amdgcn_wmma_f32_16x16x4_f32(false, a0, false, b0,
                                                   (short)0, acc0, false, false);
      a1.x = ldsA[kr0 + sm1 * 16 + ml]; a1.y = ldsA[kr1 + sm1 * 16 + ml];
      b1.x = ldsB[kr0 + sn1 * 16 + ml]; b1.y = ldsB[kr1 + sn1 * 16 + ml];
      acc1 = __builtin_amdgcn_wmma_f32_16x16x4_f32(false, a1, false, b1,
                                                   (short)0, acc1, false, false);
    }
    __syncthreads();
  }

  float ss = 0.f;
#pragma unroll
  for (int rr = 0; rr < 8; ++rr)
    ss += acc0[rr] * acc0[rr] + acc1[rr] * acc1[rr];
  red[tid] = ss;
  __syncthreads();
  for (int off = 128; off > 0; off >>= 1) {
    if (tid < off) red[tid] += red[tid + off];
    __syncthreads();
  }
  if (tid == 0) covPartials[blockIdx.x] = red[0];
}

// ---------------------------------------------------------------------------
// Kernel 4: finalize  out = 25*mean + sqrt(sum_sq)
// ---------------------------------------------------------------------------
__global__ __launch_bounds__(256) void finalize_kernel(
    const float* __restrict__ ws_ro, float* __restrict__ out) {
  __shared__ float red[256];
  const int tid = threadIdx.x;
  float s = 0.f;
  for (int i = tid; i < REPR_BLOCKS; i += 256) s += ws_ro[REPR_OFF + i];
  red[tid] = s;
  __syncthreads();
  for (int off = 128; off > 0; off >>= 1) {
    if (tid < off) red[tid] += red[tid + off];
    __syncthreads();
  }
  const float reprSum = red[0];
  __syncthreads();
  float c = 0.f;
  for (int i = tid; i < COV_BLOCKS; i += 256) c += ws_ro[COV_OFF + i];
  red[tid] = c;
  __syncthreads();
  for (int off = 128; off > 0; off >>= 1) {
    if (tid < off) red[tid] += red[tid + off];
    __syncthreads();
  }
  if (tid == 0) {
    const float repr = reprSum / (float)((size_t)N_ROWS * DIM);
    out[0] = 25.0f * repr + sqrtf(red[0]);
  }
}

// ---------------------------------------------------------------------------
extern "C" void kernel_launch(void* const* d_in, const int* in_sizes, int n_in,
                              void* d_out, int out_size, void* d_ws,
                              size_t ws_size, hipStream_t stream) {
  (void)in_sizes; (void)n_in; (void)out_size; (void)ws_size;
  const float* z_a = (const float*)d_in[0];
  const float* z_b = (const float*)d_in[1];
  const int* perm  = (const int*)d_in[2];
  float* ws  = (float*)d_ws;
  float* out = (float*)d_out;

  repr_partial_kernel<<<REPR_BLOCKS, 256, 0, stream>>>(z_a, z_b, ws + REPR_OFF);
  dim3 gGram(64, 2, KSPLIT);          // 512 workgroups for the dominant phase
  gram_kernel<<<gGram, 256, 0, stream>>>(z_a, z_b, perm, ws);
  covdiff_kernel<<<COV_BLOCKS, 256, 0, stream>>>(ws, ws + COV_OFF);
  finalize_kernel<<<1, 256, 0, stream>>>(ws, out);
}